// TopkR_12670153523748
// MI455X (gfx1250) — compile-verified
//
#include <hip/hip_runtime.h>

typedef __attribute__((ext_vector_type(2))) float v2f;
typedef __attribute__((ext_vector_type(8))) float v8f;

#define QK      128
#define NQ      1024
#define NBATCH  32
#define NTILES  64          // 1024 keys / 16 per tile
#define KPAD    132         // 128 + 4 pad -> conflict-free LDS reads
#define SCALE   0.08838834764831845f   // 128^-0.5
#define NEG_BIG (-1.0e30f)

// descending compare-exchange on (value,index); tie -> lower index first (matches jax top_k)
__device__ __forceinline__ void cswap(float& av, int& ai, float& bv, int& bi) {
    bool sw = (bv > av) || ((bv == av) && (bi < ai));
    float nv = sw ? bv : av;  float mv = sw ? av : bv;
    int   ni = sw ? bi : ai;  int   mi = sw ? ai : bi;
    av = nv; bv = mv; ai = ni; bi = mi;
}

__global__ void __launch_bounds__(256)
topk_router_kernel(const float* __restrict__ Qg,
                   const float* __restrict__ Kg,
                   float* __restrict__ Wout,
                   float* __restrict__ Iout) {
    __shared__ float lk[2][16 * KPAD];

    const int chunk = blockIdx.x;       // 0..7 : 128-query slab
    const int n     = blockIdx.y;       // batch
    const int tid   = threadIdx.x;
    const int lane  = tid & 31;
    const int wave  = tid >> 5;         // 0..7 : 16-query tile within slab
    const int half  = lane >> 4;        // lane group (selects K-pair / key half)
    const int l16   = lane & 15;

    const float* Qb = Qg + (size_t)n * NQ * QK;
    const float* Kb = Kg + (size_t)n * NQ * QK;

    // this lane's query column (B-matrix column N = l16)
    const int qrow = chunk * 128 + wave * 16 + l16;

    // ---- load Q fragments for all 32 K-steps, pre-scaled; resident in VGPRs ----
    // B layout (4x16 f32): lane L, vgpr v -> B[K = 2*(L/16)+v][N = L%16]
    v2f bfrag[32];
    {
        const float* qp = Qb + (size_t)qrow * QK + half * 2;
#pragma unroll
        for (int s = 0; s < 32; ++s) {
            float2 t = *(const float2*)(qp + 4 * s);
            bfrag[s][0] = t.x * SCALE;
            bfrag[s][1] = t.y * SCALE;
        }
    }

    // ---- cooperative K-tile staging: 256 threads x 8 floats = 16x128 tile ----
    const int ldr = tid >> 4;           // row 0..15
    const int ldc = (tid & 15) * 8;     // col 0..120

    {   // preload tile 0
        const float* kp = Kb + (size_t)ldr * QK + ldc;
        float4 r0 = *(const float4*)(kp);
        float4 r1 = *(const float4*)(kp + 4);
        *(float4*)(&lk[0][ldr * KPAD + ldc])     = r0;
        *(float4*)(&lk[0][ldr * KPAD + ldc + 4]) = r1;
    }
    __syncthreads();

    // streaming per-lane top-4 (sorted descending)
    float tv0 = NEG_BIG, tv1 = NEG_BIG, tv2 = NEG_BIG, tv3 = NEG_BIG;
    int   ti0 = 0, ti1 = 0, ti2 = 0, ti3 = 0;

    for (int t = 0; t < NTILES; ++t) {
        const int cur = t & 1;
        const int nxt = cur ^ 1;

        // prefetch next K tile from global while computing on current
        float4 p0, p1;
        if (t + 1 < NTILES) {
            const float* kp = Kb + (size_t)((t + 1) * 16 + ldr) * QK + ldc;
            p0 = *(const float4*)(kp);
            p1 = *(const float4*)(kp + 4);
        }

        // ---- stage ALL 32 A-fragments into distinct VGPRs first (16 ds ops,
        //      one wait), so the WMMA chain below runs back-to-back on the XDL
        //      pipe with no per-pair s_wait_dscnt stalls ----
        // A layout (16x4 f32): lane L, vgpr v -> A[M = L%16][K = 2*(L/16)+v]
        v2f afrag[32];
        {
            const float* la = &lk[cur][l16 * KPAD + half * 2];
#pragma unroll
            for (int s = 0; s < 32; ++s)
                afrag[s] = *(const v2f*)(la + 4 * s);
        }

        // ---- 16x16 logit tile: D = K_tile x Q_tile^T, K reduced over 128 ----
        v8f acc = {0.f, 0.f, 0.f, 0.f, 0.f, 0.f, 0.f, 0.f};
#pragma unroll
        for (int s = 0; s < 32; ++s) {
            acc = __builtin_amdgcn_wmma_f32_16x16x4_f32(
                      false, afrag[s], false, bfrag[s], (short)0, acc, false, false);
        }

        // commit prefetched tile to the other LDS buffer (independent of acc;
        // issues before the top-k VALU work so the store latency hides there)
        if (t + 1 < NTILES) {
            *(float4*)(&lk[nxt][ldr * KPAD + ldc])     = p0;
            *(float4*)(&lk[nxt][ldr * KPAD + ldc + 4]) = p1;
        }

        // ---- branchless top-4 insertion of this lane's 8 keys ----
        // C/D layout: lanes 0-15 vgpr v -> key row v; lanes 16-31 -> row v+8
#pragma unroll
        for (int v = 0; v < 8; ++v) {
            float c  = acc[v];
            int   ci = t * 16 + half * 8 + v;
            // keys arrive in ascending index order -> strict '>' keeps stable ties
            bool  rep = c > tv3;
            float nv  = rep ? c  : tv3;
            int   ni  = rep ? ci : ti3;
            bool  s2  = nv > tv2;
            float a2v = s2 ? nv  : tv2;  tv3 = s2 ? tv2 : nv;
            int   a2i = s2 ? ni  : ti2;  ti3 = s2 ? ti2 : ni;
            bool  s1  = a2v > tv1;
            float a1v = s1 ? a2v : tv1;  tv2 = s1 ? tv1 : a2v;
            int   a1i = s1 ? a2i : ti1;  ti2 = s1 ? ti1 : a2i;
            bool  s0  = a1v > tv0;
            tv1 = s0 ? tv0 : a1v;        ti1 = s0 ? ti0 : a1i;
            tv0 = s0 ? a1v : tv0;        ti0 = s0 ? a1i : ti0;
        }

        __syncthreads();
    }

    // ---- merge partner lane's top-4 (lane xor 16 holds the other key half) ----
    float bv0 = __shfl_xor(tv0, 16);  int bi0 = __shfl_xor(ti0, 16);
    float bv1 = __shfl_xor(tv1, 16);  int bi1 = __shfl_xor(ti1, 16);
    float bv2 = __shfl_xor(tv2, 16);  int bi2 = __shfl_xor(ti2, 16);
    float bv3 = __shfl_xor(tv3, 16);  int bi3 = __shfl_xor(ti3, 16);

    // bitonic: desc list ++ asc-reversed desc list; first stage keeps the 4 largest
    cswap(tv0, ti0, bv3, bi3);
    cswap(tv1, ti1, bv2, bi2);
    cswap(tv2, ti2, bv1, bi1);
    cswap(tv3, ti3, bv0, bi0);
    // sort the bitonic top-4 descending
    cswap(tv0, ti0, tv2, ti2);
    cswap(tv1, ti1, tv3, ti3);
    cswap(tv0, ti0, tv1, ti1);
    cswap(tv2, ti2, tv3, ti3);

    // ---- softmax over the 4 selected logits ----
    float m  = tv0;
    float e0 = __expf(tv0 - m);
    float e1 = __expf(tv1 - m);
    float e2 = __expf(tv2 - m);
    float e3 = __expf(tv3 - m);
    float inv = 1.0f / (e0 + e1 + e2 + e3);

    if (half == 0) {
        size_t base = ((size_t)n * NQ + qrow) * 4;
        float4 w  = make_float4(e0 * inv, e1 * inv, e2 * inv, e3 * inv);
        float4 ix = make_float4((float)ti0, (float)ti1, (float)ti2, (float)ti3);
        *(float4*)(Wout + base) = w;
        *(float4*)(Iout + base) = ix;
    }
}

extern "C" void kernel_launch(void* const* d_in, const int* in_sizes, int n_in,
                              void* d_out, int out_size, void* d_ws, size_t ws_size,
                              hipStream_t stream) {
    const float* Qg = (const float*)d_in[0];
    const float* Kg = (const float*)d_in[1];
    float* Wout = (float*)d_out;                                   // (32,1024,4) weights
    float* Iout = (float*)d_out + (size_t)NBATCH * NQ * 4;         // (32,1024,4) indices

    dim3 grid(NQ / 128, NBATCH);   // (8, 32): 128-query slab per block
    dim3 block(256);               // 8 waves, one 16-query tile each
    topk_router_kernel<<<grid, block, 0, stream>>>(Qg, Kg, Wout, Iout);
}